// ProposalGenerator_2181843386585
// MI455X (gfx1250) — compile-verified
//
#include <hip/hip_runtime.h>
#include <hip/hip_bf16.h>
#include <stdint.h>

#define PRE_TOPK   6000
#define POST_TOPK  1000
#define NMS_THR    0.7f
#define MIN_SZ     16.0f
#define IMG_H      800.0f
#define IMG_W      1216.0f
#define NBINS      (1u << 17)   // histogram over top-17 bits of sortable key
#define CAND_CAP   8192
#define NTOP_PAD   6016         // 6000 padded to multiple of 32 (and of 4 keys / 16B)
#define NW         188          // ceil(6000/32) suppression-mask words per row

typedef unsigned u32x4 __attribute__((ext_vector_type(4)));
typedef unsigned u32x8 __attribute__((ext_vector_type(8)));

// ---------------------------------------------------------------------------
// CDNA5 helpers.
// Low 32 bits of a flat pointer to __shared__ == workgroup-relative LDS offset
// (flat->LDS mapping truncates to addr[31:0], ISA 10.2).
// ---------------------------------------------------------------------------
static __device__ __forceinline__ unsigned lds_off(const void* p) {
  return (unsigned)(unsigned long long)p;
}
static __device__ __forceinline__ void async_copy_b128(unsigned lds, const void* g) {
  asm volatile("global_load_async_to_lds_b128 %0, %1, off"
               :: "v"(lds), "v"(g) : "memory");
}
static __device__ __forceinline__ void wait_async0() {
  asm volatile("s_wait_asynccnt 0" ::: "memory");
}
// One-shot TDM DMA: contiguous `words` DWORDs from global -> LDS.
// D# per CDNA5 ISA (2 SGPR groups, 1-row 2D tile).
static __device__ __forceinline__ void tdm_load_dwords(unsigned lds, const void* gp,
                                                       unsigned words) {
  unsigned long long ga = (unsigned long long)gp;
  u32x4 g0;
  g0[0] = 1u;                                            // count=1 (valid), no gather
  g0[1] = lds;                                           // lds_addr (bytes)
  g0[2] = (unsigned)ga;                                  // global_addr[31:0]
  g0[3] = (unsigned)((ga >> 32) & 0x01FFFFFFu)           // global_addr[56:32]
        | (2u << 30);                                    // type=2 ("image")
  u32x8 g1;
  g1[0] = (2u << 16);                                    // data_size=4B, wg_mask=0
  g1[1] = (words & 0xFFFFu) << 16;                       // tensor_dim0[15:0]
  g1[2] = (words >> 16) | (1u << 16);                    // tensor_dim0[31:16] | tensor_dim1=1
  g1[3] = (words & 0xFFFFu) << 16;                       // tile_dim0 = words
  g1[4] = 1u;                                            // tile_dim1=1, tile_dim2=0
  g1[5] = words;                                         // tensor_dim0_stride[31:0]
  g1[6] = 0u;                                            // stride0 hi | stride1 lo
  g1[7] = 0u;
  asm volatile("tensor_load_to_lds %0, %1" :: "s"(g0), "s"(g1) : "memory");
  asm volatile("s_wait_tensorcnt 0x0" ::: "memory");
}
static __device__ __forceinline__ unsigned ballot32(bool p) {
#if __has_builtin(__builtin_amdgcn_ballot_w32)
  return __builtin_amdgcn_ballot_w32(p);
#else
  return (unsigned)__ballot(p);
#endif
}

// Shared decode math (used by decode pass and by the top-k gather pass).
static __device__ __forceinline__ float4 decode_box(const float4 a, const float4 d) {
  float wa = a.z - a.x, ha = a.w - a.y;
  float xa = a.x + 0.5f * wa, ya = a.y + 0.5f * ha;
  float x = d.x * wa + xa, y = d.y * ha + ya;
  float w = expf(d.z) * wa, h = expf(d.w) * ha;
  float x1 = fminf(fmaxf(x - 0.5f * w, 0.0f), IMG_W - 1.0f);
  float y1 = fminf(fmaxf(y - 0.5f * h, 0.0f), IMG_H - 1.0f);
  float x2 = fminf(fmaxf(x + 0.5f * w, 0.0f), IMG_W - 1.0f);
  float y2 = fminf(fmaxf(y + 0.5f * h, 0.0f), IMG_H - 1.0f);
  return make_float4(x1, y1, x2, y2);
}

// ---------------------------------------------------------------------------
// K1: decode boxes, build sortable keys, global histogram (memory-bound pass).
// key: monotone float->uint map; score>=0  <=>  key>=0x80000000.
// ---------------------------------------------------------------------------
__global__ void k_decode(const float4* __restrict__ anchors,
                         const float4* __restrict__ deltas,
                         const float*  __restrict__ logits,
                         unsigned* __restrict__ keys,
                         unsigned* __restrict__ hist,
                         int n) {
  int i = blockIdx.x * blockDim.x + threadIdx.x;
  if (i >= n) return;
  float4 b = decode_box(anchors[i], deltas[i]);
  bool ok = (b.z - b.x >= MIN_SZ) && (b.w - b.y >= MIN_SZ);
  float s = ok ? (1.0f / (1.0f + expf(-logits[i]))) : -1.0f;
  unsigned fb = __float_as_uint(s);
  unsigned key = (fb & 0x80000000u) ? ~fb : (fb | 0x80000000u);
  keys[i] = key;
  atomicAdd(&hist[key >> 15], 1u);
}

// ---------------------------------------------------------------------------
// K2: single block; find largest bin B s.t. count(key>>15 >= B) >= PRE_TOPK.
// ---------------------------------------------------------------------------
__global__ __launch_bounds__(1024) void k_select_bin(const unsigned* __restrict__ hist,
                                                     unsigned* __restrict__ cnt) {
  __shared__ unsigned part[1024];
  __shared__ unsigned csum[128];
  __shared__ unsigned sbuf[1024];
  __shared__ int selChunk;
  __shared__ unsigned beforeS;
  const int t = threadIdx.x;
  {                                     // all 1024 threads: 128 contiguous bins each
    unsigned s = 0;
    const unsigned* h = hist + (size_t)t * 128;
    for (int k = 0; k < 128; ++k) s += h[k];
    part[t] = s;
  }
  __syncthreads();
  if (t < 128) {                        // chunk (1024-bin) sums
    unsigned s = 0;
    for (int k = 0; k < 8; ++k) s += part[t * 8 + k];
    csum[t] = s;
  }
  __syncthreads();
  if (t == 0) {
    unsigned acc = 0;
    int c = 127;
    for (; c >= 0; --c) {
      if (acc + csum[c] >= PRE_TOPK) break;
      acc += csum[c];
    }
    if (c < 0) c = 0;
    selChunk = c;
    beforeS = acc;
  }
  __syncthreads();
  sbuf[t] = hist[(size_t)selChunk * 1024 + t];
  __syncthreads();
  for (int d = 1; d < 1024; d <<= 1) {  // inclusive suffix sum
    unsigned add = (t + d < 1024) ? sbuf[t + d] : 0u;
    __syncthreads();
    sbuf[t] += add;
    __syncthreads();
  }
  unsigned before = beforeS;
  unsigned nxt = (t == 1023) ? 0u : sbuf[t + 1];
  if (before + sbuf[t] >= PRE_TOPK && before + nxt < PRE_TOPK)
    cnt[0] = (unsigned)selChunk * 1024u + (unsigned)t;
}

// ---------------------------------------------------------------------------
// K3: compact all candidates with (key>>15) >= B  (>=6000 of them, incl. ties).
// ---------------------------------------------------------------------------
__global__ void k_compact(const unsigned* __restrict__ keys,
                          unsigned* __restrict__ cnt,
                          unsigned* __restrict__ candk,
                          int* __restrict__ candi, int n) {
  int i = blockIdx.x * blockDim.x + threadIdx.x;
  if (i >= n) return;
  unsigned key = keys[i];
  if ((key >> 15) >= cnt[0]) {
    unsigned slot = atomicAdd(&cnt[1], 1u);
    if (slot < CAND_CAP) { candk[slot] = key; candi[slot] = i; }
  }
}

// ---------------------------------------------------------------------------
// K4: single-block bitonic sort of 8192 packed (key<<32 | ~idx) in 64KB LDS,
// descending => key desc, index asc (lax.top_k tie-break). Then gather &
// re-decode the top-6000 boxes, zero-pad to NTOP_PAD.
// ---------------------------------------------------------------------------
__global__ __launch_bounds__(1024) void k_sort_top(const unsigned* __restrict__ candk,
                                                   const int* __restrict__ candi,
                                                   const unsigned* __restrict__ cnt,
                                                   const float4* __restrict__ anchors,
                                                   const float4* __restrict__ deltas,
                                                   float4* __restrict__ topb,
                                                   unsigned* __restrict__ topk) {
  __shared__ unsigned long long sk[CAND_CAP];   // 64 KB of the 320 KB/WGP LDS
  const int t = threadIdx.x;
  unsigned count = cnt[1];
  if (count > CAND_CAP) count = CAND_CAP;
  for (int i = t; i < CAND_CAP; i += 1024) {
    unsigned long long v = 0ull;
    if ((unsigned)i < count)
      v = ((unsigned long long)candk[i] << 32) | (unsigned)(~(unsigned)candi[i]);
    sk[i] = v;
  }
  __syncthreads();
  for (unsigned k = 2; k <= CAND_CAP; k <<= 1) {
    for (unsigned j = k >> 1; j > 0; j >>= 1) {
      for (unsigned i = t; i < CAND_CAP; i += 1024) {
        unsigned ixj = i ^ j;
        if (ixj > i) {
          unsigned long long a = sk[i], b = sk[ixj];
          bool up = ((i & k) == 0);
          bool sw = up ? (a < b) : (a > b);   // descending overall
          if (sw) { sk[i] = b; sk[ixj] = a; }
        }
      }
      __syncthreads();
    }
  }
  for (int s = t; s < NTOP_PAD; s += 1024) {
    float4 p = make_float4(0.f, 0.f, 0.f, 0.f);
    unsigned key = 0u;
    if (s < PRE_TOPK) {
      unsigned long long v = sk[s];
      key = (unsigned)(v >> 32);
      if (key) {
        unsigned idx = ~(unsigned)(v & 0xFFFFFFFFull);
        p = decode_box(anchors[idx], deltas[idx]);
      }
    }
    topb[s] = p;
    topk[s] = key;
  }
}

// ---------------------------------------------------------------------------
// K5: pairwise suppression bitmask. One wave per 32(rows) x 32(cols) tile.
// Row boxes staged via per-lane async global->LDS b128 (ASYNCcnt path);
// one wave32 ballot per row -> one mask word.
// mask[i*NW + w] bit b : box (w*32+b) has IoU>0.7 with box i (and index > i).
// ---------------------------------------------------------------------------
__global__ void k_nms_mask(const float4* __restrict__ topb,
                           unsigned* __restrict__ mask) {
  __shared__ float4 rowb[32];
  const int lane  = threadIdx.x;        // 0..31, wave32
  const int jw    = blockIdx.x;         // column word
  const int j     = jw * 32 + lane;
  const int ibase = blockIdx.y * 32;

  async_copy_b128(lds_off(&rowb[lane]), (const void*)&topb[ibase + lane]);
  float4 bj = topb[j];
  float aj = (bj.z - bj.x) * (bj.w - bj.y);
  wait_async0();
  __syncthreads();

#pragma unroll 4
  for (int ii = 0; ii < 32; ++ii) {
    const int i = ibase + ii;
    float4 bi = rowb[ii];
    float ai  = (bi.z - bi.x) * (bi.w - bi.y);
    float xx1 = fmaxf(bi.x, bj.x);
    float yy1 = fmaxf(bi.y, bj.y);
    float xx2 = fminf(bi.z, bj.z);
    float yy2 = fminf(bi.w, bj.w);
    float inter = fmaxf(xx2 - xx1, 0.f) * fmaxf(yy2 - yy1, 0.f);
    bool sup = (j > i) && (j < PRE_TOPK) && (inter > NMS_THR * (ai + aj - inter));
    unsigned w = ballot32(sup);
    if (lane == 0) mask[(size_t)i * NW + jw] = w;
  }
}

// ---------------------------------------------------------------------------
// K6: single-block greedy scan. The 24 KB sorted-key array is staged into LDS
// with ONE Tensor Data Mover DMA (tensor_load_to_lds + s_wait_tensorcnt),
// issued by wave 0. Thread 0 serially skips dead rows (LDS reads); only
// ~1000 barrier rounds. Writes the (1000,4) output, zeros for unkept slots.
// ---------------------------------------------------------------------------
__global__ __launch_bounds__(256) void k_nms_greedy(const unsigned* __restrict__ topk,
                                                    const float4* __restrict__ topb,
                                                    const unsigned* __restrict__ mask,
                                                    float4* __restrict__ out) {
  __shared__ unsigned skey[NTOP_PAD];
  __shared__ unsigned sup[NW];
  __shared__ int keepbuf[POST_TOPK];
  __shared__ int nkeepS;
  __shared__ int curS;
  const int tid = threadIdx.x;

  if (tid < 32)                                  // wave 0: one TDM DMA for all keys
    tdm_load_dwords(lds_off(&skey[0]), (const void*)topk, NTOP_PAD);
  for (int w = tid; w < NW; w += 256) sup[w] = 0u;
  if (tid == 0) { nkeepS = 0; curS = -1; }
  __syncthreads();

  int i = 0;
  while (true) {
    if (tid == 0) {
      int f = -1;
      while (i < PRE_TOPK && nkeepS < POST_TOPK) {
        if ((skey[i] & 0x80000000u) && !((sup[i >> 5] >> (i & 31)) & 1u)) { f = i; break; }
        ++i;
      }
      curS = f;
      if (f >= 0) { keepbuf[nkeepS] = f; nkeepS = nkeepS + 1; ++i; }
    }
    __syncthreads();
    int f = curS;
    if (f < 0) break;
    for (int w = tid; w < NW; w += 256) sup[w] |= mask[(size_t)f * NW + w];
    __syncthreads();
  }

  int nk = nkeepS;
  for (int k = tid; k < POST_TOPK; k += 256) {
    float4 v = make_float4(0.f, 0.f, 0.f, 0.f);
    if (k < nk) v = topb[keepbuf[k]];
    out[k] = v;
  }
}

// ---------------------------------------------------------------------------
// Host side: ws layout + launches (graph-capture safe: only async ops).
// Inputs: [0]=images, [1]=origin_sizes, [2]=anchors(N,4), [3]=deltas(N,4),
//         [4]=logits(N,1).  Output: (1000,4) float.
// ---------------------------------------------------------------------------
extern "C" void kernel_launch(void* const* d_in, const int* in_sizes, int n_in,
                              void* d_out, int out_size, void* d_ws, size_t ws_size,
                              hipStream_t stream) {
  (void)n_in; (void)out_size; (void)ws_size;
  const float4* anchors = (const float4*)d_in[2];
  const float4* deltas  = (const float4*)d_in[3];
  const float*  logits  = (const float*)d_in[4];
  const int n = in_sizes[2] / 4;

  char* ws = (char*)d_ws;
  size_t off = 0;
  auto alloc = [&](size_t bytes) -> size_t {
    size_t o = off;
    off += (bytes + 255) & ~(size_t)255;
    return o;
  };
  size_t o_keys = alloc((size_t)n * 4);
  size_t o_hist = alloc((size_t)NBINS * 4);
  size_t o_cnt  = alloc(256);
  size_t o_ck   = alloc((size_t)CAND_CAP * 4);
  size_t o_ci   = alloc((size_t)CAND_CAP * 4);
  size_t o_tb   = alloc((size_t)NTOP_PAD * 16);
  size_t o_tk   = alloc((size_t)NTOP_PAD * 4);
  size_t o_mask = alloc((size_t)NTOP_PAD * NW * 4);

  unsigned* keys  = (unsigned*)(ws + o_keys);
  unsigned* hist  = (unsigned*)(ws + o_hist);
  unsigned* cnt   = (unsigned*)(ws + o_cnt);
  unsigned* candk = (unsigned*)(ws + o_ck);
  int*      candi = (int*)(ws + o_ci);
  float4*   topb  = (float4*)(ws + o_tb);
  unsigned* topk  = (unsigned*)(ws + o_tk);
  unsigned* mask  = (unsigned*)(ws + o_mask);

  hipMemsetAsync(hist, 0, (size_t)NBINS * 4, stream);
  hipMemsetAsync(cnt, 0, 256, stream);

  const int blk = 256;
  const int grid = (n + blk - 1) / blk;
  k_decode<<<grid, blk, 0, stream>>>(anchors, deltas, logits, keys, hist, n);
  k_select_bin<<<1, 1024, 0, stream>>>(hist, cnt);
  k_compact<<<grid, blk, 0, stream>>>(keys, cnt, candk, candi, n);
  k_sort_top<<<1, 1024, 0, stream>>>(candk, candi, cnt, anchors, deltas, topb, topk);
  dim3 gmask(NW, NTOP_PAD / 32, 1);
  k_nms_mask<<<gmask, 32, 0, stream>>>(topb, mask);
  k_nms_greedy<<<1, 256, 0, stream>>>(topk, topb, mask, (float4*)d_out);
}